// E3nnModel_10110353015096
// MI455X (gfx1250) — compile-verified
//
#include <hip/hip_runtime.h>

typedef __attribute__((ext_vector_type(2))) float v2f;
typedef __attribute__((ext_vector_type(8))) float v8f;

__device__ __forceinline__ float rfl(float x) {
    union { float f; int i; } u;
    u.f = x;
    u.i = __builtin_amdgcn_readfirstlane(u.i);
    return u.f;
}

// -----------------------------------------------------------------------------
// Fold kernel (1 wave): c = (w_tp / sqrt(12)) * W1 @ W2 @ (W3 @ ones(3))
// Uses V_WMMA_F32_16X16X4_F32 twice for the two K=2 mat-vecs, with a
// layout-robust B-operand fill (K2/K3 contributions are zeroed via A).
// -----------------------------------------------------------------------------
__global__ __launch_bounds__(32) void fold_weights_kernel(
    const float* __restrict__ w_tp,   // scalar
    const float* __restrict__ W1,     // [3,2] row-major
    const float* __restrict__ W2,     // [2,2] row-major
    const float* __restrict__ W3,     // [2,3] row-major
    float* __restrict__ c_out)        // [3]
{
    const int  lane = threadIdx.x;     // 0..31
    const int  m    = lane & 15;       // A-matrix row (doc layout: lane%16)
    const bool lo   = lane < 16;       // lanes 0-15 hold K=0,1; 16-31 hold K=2,3

    // v3 = W3 @ ones(3)  (uniform across lanes)
    const float v3_0 = W3[0] + W3[1] + W3[2];
    const float v3_1 = W3[3] + W3[4] + W3[5];

    // ---- WMMA #1: D1[m][n] = sum_{k<2} W2[m][k] * v3[k] ----
    // A (16x4): rows 0-1 = W2, all K=2,3 slots (lanes 16-31) zeroed.
    v2f a1;
    a1.x = (lo && m < 2) ? W2[2 * m + 0] : 0.0f;
    a1.y = (lo && m < 2) ? W2[2 * m + 1] : 0.0f;
    // B (4x16): every plausible K=0 slot gets v3_0, every K=1 slot gets v3_1;
    // K=2/3 slots are don't-care (multiplied by zeroed A halves).
    v2f b1;
    b1.x = lo ? v3_0 : v3_1;
    b1.y = v3_1;
    v8f d1 = {};
    d1 = __builtin_amdgcn_wmma_f32_16x16x4_f32(false, a1, false, b1,
                                               (short)0, d1, false, false);
    // C/D layout: VGPR r, lane 0 -> (M=r, N=0)
    const float v2_0 = rfl(d1[0]);
    const float v2_1 = rfl(d1[1]);

    // ---- WMMA #2: D2[m][n] = sum_{k<2} W1[m][k] * v2[k] ----
    v2f a2;
    a2.x = (lo && m < 3) ? W1[2 * m + 0] : 0.0f;
    a2.y = (lo && m < 3) ? W1[2 * m + 1] : 0.0f;
    v2f b2;
    b2.x = lo ? v2_0 : v2_1;
    b2.y = v2_1;
    v8f d2 = {};
    d2 = __builtin_amdgcn_wmma_f32_16x16x4_f32(false, a2, false, b2,
                                               (short)0, d2, false, false);

    // scale = w_tp * 1/sqrt(3) * 1/sqrt(2) * 1/sqrt(2) = w_tp / sqrt(12)
    const float scale = w_tp[0] * 0.28867513459481287f;
    const float c0 = rfl(d2[0]) * scale;
    const float c1 = rfl(d2[1]) * scale;
    const float c2 = rfl(d2[2]) * scale;
    if (lane == 0) {
        c_out[0] = c0;
        c_out[1] = c1;
        c_out[2] = c2;
    }
}

// -----------------------------------------------------------------------------
// Streaming kernel: out[i] = z[i] * dot(pos[i], c).  4 atoms per thread via
// three float4 (b128) loads of pos (contiguous 48B per lane -> wave covers a
// dense 1536B region), one int4 load of z, one float4 store.  Memory-bound.
// -----------------------------------------------------------------------------
__global__ __launch_bounds__(256) void scale_dot_kernel(
    const float* __restrict__ pos,   // [N,3]
    const int* __restrict__ z,       // [N]
    const float* __restrict__ c,     // [3] folded weights
    float* __restrict__ out,         // [N]
    int n)
{
    const float c0 = c[0], c1 = c[1], c2 = c[2];

    const int q  = blockIdx.x * blockDim.x + threadIdx.x;  // quad index
    const int nq = n >> 2;                                 // full quads

    if (q < nq) {
        const float4* p4 = reinterpret_cast<const float4*>(pos) + (size_t)q * 3;
        const float4 pa = p4[0];  // a0.x a0.y a0.z a1.x
        const float4 pb = p4[1];  // a1.y a1.z a2.x a2.y
        const float4 pc = p4[2];  // a2.z a3.x a3.y a3.z
        const int4   zz = reinterpret_cast<const int4*>(z)[q];

        float4 r;
        r.x = (float)zz.x * (pa.x * c0 + pa.y * c1 + pa.z * c2);
        r.y = (float)zz.y * (pa.w * c0 + pb.x * c1 + pb.y * c2);
        r.z = (float)zz.z * (pb.z * c0 + pb.w * c1 + pc.x * c2);
        r.w = (float)zz.w * (pc.y * c0 + pc.z * c1 + pc.w * c2);
        reinterpret_cast<float4*>(out)[q] = r;
    }

    // Tail (n not divisible by 4): first few threads also handle one atom each.
    const int rem = n & 3;
    if (q < rem) {
        const int i = nq * 4 + q;
        out[i] = (float)z[i] *
                 (pos[3 * (size_t)i + 0] * c0 +
                  pos[3 * (size_t)i + 1] * c1 +
                  pos[3 * (size_t)i + 2] * c2);
    }
}

extern "C" void kernel_launch(void* const* d_in, const int* in_sizes, int n_in,
                              void* d_out, int out_size, void* d_ws, size_t ws_size,
                              hipStream_t stream) {
    const float* pos = (const float*)d_in[0];
    const int*   zin = (const int*)d_in[1];
    const float* wtp = (const float*)d_in[2];
    const float* W1  = (const float*)d_in[3];
    const float* W2  = (const float*)d_in[4];
    const float* W3  = (const float*)d_in[5];
    float* out  = (float*)d_out;
    float* cvec = (float*)d_ws;   // 3 floats of scratch

    const int n = in_sizes[1];    // N atoms (z element count)

    fold_weights_kernel<<<1, 32, 0, stream>>>(wtp, W1, W2, W3, cvec);

    const int nq      = n >> 2;
    const int threads = 256;
    int blocks = (nq + threads - 1) / threads;
    if (blocks < 1) blocks = 1;
    scale_dot_kernel<<<blocks, threads, 0, stream>>>(pos, zin, cvec, out, n);
}